// MultiLabelLoss_14319420965549
// MI455X (gfx1250) — compile-verified
//
#include <hip/hip_runtime.h>
#include <math.h>

// Cross-entropy over B=4096 rows of C=50257 logits, T=8 labels/row (-1 = ignore).
// loss_b = logsumexp(x[b,:]) - mean_t(x[b, y[b,t]]);  output = mean_b(loss_b).
//
// Single-pass online softmax per row: x (823 MB) is read from HBM exactly once
// -> ~35us floor at 23.3 TB/s. One 256-thread block (8 wave32) per row, 4096
// blocks. Software-pipelined inner loop: next global_load_b128 is issued before
// the current float4's v_exp_f32 work, so each wave keeps a load in flight
// while the TRANS pipe works; global_prefetch_b8 streams ~32KB ahead into GL2.
// Deterministic final reduction (no float atomics) -> graph replays bit-stable.

#define BDIM 256
constexpr int kC = 50257;
constexpr int kT = 8;

// Online-softmax pair combine: (m,s) <- combine((m,s),(m2,s2)).
// Guarded so (-inf, 0) neutral elements never produce NaN.
__device__ __forceinline__ void ol_combine(float& m, float& s, float m2, float s2) {
  float M  = fmaxf(m, m2);
  float e1 = (m  > -INFINITY) ? __expf(m  - M) : 0.0f;
  float e2 = (m2 > -INFINITY) ? __expf(m2 - M) : 0.0f;
  s = s * e1 + s2 * e2;
  m = M;
}

__global__ __launch_bounds__(BDIM) void row_loss_kernel(
    const float* __restrict__ x, const int* __restrict__ y,
    float* __restrict__ row_loss, int B) {
  const int b   = blockIdx.x;
  const int tid = threadIdx.x;
  const float* __restrict__ xrow = x + (size_t)b * kC;

  // Rows are only dword-aligned (C % 4 != 0); CDNA5 global b128 loads accept
  // dword alignment. Strided float4 across the block -> each wave issue pulls
  // 512B contiguous.
  const float4* __restrict__ row4 = reinterpret_cast<const float4*>(xrow);
  const int n4 = kC >> 2;  // 12564 float4s + 1 scalar tail element

  float m  = -INFINITY;
  float sa = 0.0f;   // two independent accumulators: shorter add chain behind
  float sb = 0.0f;   // the v_exp_f32 results

  // Software-pipelined main loop (n4 > BDIM, so every thread has >= 1 iter).
  int i = tid;
  float4 cur = row4[i];
  i += BDIM;
  while (true) {
    const bool have_next = (i < n4);
    float4 nxt;
    if (have_next) nxt = row4[i];            // next load in flight now
    __builtin_prefetch(row4 + i + BDIM * 8, 0, 0);  // gfx1250 global_prefetch_b8

    float c = fmaxf(fmaxf(cur.x, cur.y), fmaxf(cur.z, cur.w));
    if (c > m) {               // rare after warm-up -> ~1 exp/element total
      float e = __expf(m - c); // m=-inf first time: exp(-inf)=0 zeroes sums
      sa *= e;
      sb *= e;
      m = c;
    }
    sa += __expf(cur.x - m) + __expf(cur.y - m);
    sb += __expf(cur.z - m) + __expf(cur.w - m);

    if (!have_next) break;
    cur = nxt;
    i += BDIM;
  }
  float s = sa + sb;

  // Scalar tail (1 element: index 50256, handled by thread 0 only).
  for (int t = (n4 << 2) + tid; t < kC; t += BDIM) {
    float v = xrow[t];
    if (v > m) { s *= __expf(m - v); m = v; }
    s += __expf(v - m);
  }

  // Intra-wave (wave32) butterfly combine of (m, s).
  #pragma unroll
  for (int off = 16; off > 0; off >>= 1) {
    float m2 = __shfl_xor(m, off, 32);
    float s2 = __shfl_xor(s, off, 32);
    ol_combine(m, s, m2, s2);
  }

  // Cross-wave combine through LDS (8 waves per block).
  __shared__ float sm[BDIM / 32];
  __shared__ float ss[BDIM / 32];
  const int wave = tid >> 5;
  const int lane = tid & 31;
  if (lane == 0) { sm[wave] = m; ss[wave] = s; }
  __syncthreads();

  if (wave == 0) {
    m = (lane < BDIM / 32) ? sm[lane] : -INFINITY;
    s = (lane < BDIM / 32) ? ss[lane] : 0.0f;
    #pragma unroll
    for (int off = 16; off > 0; off >>= 1) {
      float m2 = __shfl_xor(m, off, 32);
      float s2 = __shfl_xor(s, off, 32);
      ol_combine(m, s, m2, s2);
    }
    if (lane == 0) {
      float lse = m + __logf(s);
      const int* __restrict__ yrow = y + b * kT;
      int cnt = 0;
      float g = 0.0f;
      #pragma unroll
      for (int t = 0; t < kT; ++t) {
        int idx = yrow[t];
        if (idx != -1) { ++cnt; g += xrow[idx]; }
      }
      row_loss[b] = (cnt > 0) ? (lse - g / (float)cnt) : 0.0f;
    }
  }
}

// Deterministic mean of B per-row losses (single block, fixed tree order).
__global__ __launch_bounds__(BDIM) void mean_kernel(
    const float* __restrict__ v, float* __restrict__ out, int n) {
  const int tid = threadIdx.x;
  float s = 0.0f;
  for (int i = tid; i < n; i += BDIM) s += v[i];
  #pragma unroll
  for (int off = 16; off > 0; off >>= 1) s += __shfl_xor(s, off, 32);

  __shared__ float ws[BDIM / 32];
  const int wave = tid >> 5;
  const int lane = tid & 31;
  if (lane == 0) ws[wave] = s;
  __syncthreads();
  if (tid == 0) {
    float t = 0.0f;
    #pragma unroll
    for (int w = 0; w < BDIM / 32; ++w) t += ws[w];
    out[0] = t / (float)n;
  }
}

extern "C" void kernel_launch(void* const* d_in, const int* in_sizes, int n_in,
                              void* d_out, int out_size, void* d_ws, size_t ws_size,
                              hipStream_t stream) {
  const float* x = (const float*)d_in[0];  // [B, C] float32
  const int*   y = (const int*)d_in[1];    // [B, T] int labels (-1 = ignore)
  float* out = (float*)d_out;              // scalar float32
  float* ws  = (float*)d_ws;               // B floats of scratch

  const int B = in_sizes[0] / kC;          // 4096

  row_loss_kernel<<<B, BDIM, 0, stream>>>(x, y, ws, B);
  mean_kernel<<<1, BDIM, 0, stream>>>(ws, out, B);
}